// ChiENNAggregate_86139864089509
// MI455X (gfx1250) — compile-verified
//
#include <hip/hip_runtime.h>
#include <hip/hip_bf16.h>

typedef __attribute__((ext_vector_type(2))) float v2f;
typedef __attribute__((ext_vector_type(4))) float v4f;
typedef __attribute__((ext_vector_type(8))) float v8f;

#define H 128
#define NBR 6
#define WAVES_PER_BLOCK 4
#define ROWS_PER_WAVE 32          // two 16-row WMMA M-tiles per wave
#define MT 2                      // M-tiles per wave

// V_WMMA_F32_16X16X4_F32: D(16x16) += A(16x4) * B(4x16), wave32. K=128 -> 32 k-steps.
// A layout: lane l(0..15)+16h holds row M=l, K = 4*ks + 2h + {0,1}
// B layout: lane l(0..15)+16h holds col N=l, K rows 4*ks + 2h + {0,1}
// C/D layout: vgpr v, lane l+16h -> M = v + 8h, N = l

__global__ __launch_bounds__(WAVES_PER_BLOCK * 32)
void chienn_fused_kernel(const float* __restrict__ x,
                         const float* __restrict__ msg,
                         const unsigned char* __restrict__ mask,
                         const int* __restrict__ pidx,
                         const float* __restrict__ W_self,
                         const float* __restrict__ b_self,
                         const float* __restrict__ W_par,
                         const float* __restrict__ b_par,
                         const float* __restrict__ W_post,
                         const float* __restrict__ b_post,
                         float* __restrict__ out,
                         int n)
{
    extern __shared__ float ldsbuf[];   // WAVES_PER_BLOCK * 32 * 128 floats = 64 KB

    const int lane = threadIdx.x & 31;
    const int wave = threadIdx.x >> 5;
    const int half = lane >> 4;      // 0 or 1
    const int l    = lane & 15;      // 0..15
    const int r0   = (blockIdx.x * WAVES_PER_BLOCK + wave) * ROWS_PER_WAVE;
    if (r0 >= n) return;             // wave-uniform exit

    float* agg = ldsbuf + wave * (ROWS_PER_WAVE * H);

    // ---------------- Phase 0: masked msg sum + b_self + b_par -> LDS ----------------
    // Each lane covers 4 consecutive columns; 32 lanes cover one full H=128 row.
    {
        const int c0 = lane * 4;
        v4f bsum = *(const v4f*)(b_self + c0) + *(const v4f*)(b_par + c0);
        for (int r = 0; r < ROWS_PER_WAVE; ++r) {
            int row = r0 + r; if (row > n - 1) row = n - 1;
            // speculative stream prefetch two rows ahead (global_prefetch_b8)
            {
                int prow = row + 2; if (prow > n - 1) prow = n - 1;
                const float* pf = msg + (size_t)prow * NBR * H + c0;
#pragma unroll
                for (int b = 0; b < NBR; ++b)
                    __builtin_prefetch(pf + b * H, 0, 0);
            }
            const float* mrow = msg + (size_t)row * NBR * H + c0;
            const unsigned char* mk = mask + (size_t)row * NBR;
            v4f s = bsum;
#pragma unroll
            for (int b = 0; b < NBR; ++b) {
                // mask is uniform across the wave for a given (row, b):
                // the branch skips the whole 512B msg row when masked out.
                if (mk[b]) {
                    s += __builtin_nontemporal_load((const v4f*)(mrow + b * H));
                }
            }
            *(v4f*)(agg + r * H + c0) = s;
        }
    }

    // ---------------- Phase 1: acc = x_tile @ W_self + gather(x) @ W_par ----------------
    // Two M-tiles share every B fragment -> half the W-fragment traffic per WMMA.
    int rowA0 = r0 + l;      if (rowA0 > n - 1) rowA0 = n - 1;
    int rowA1 = r0 + 16 + l; if (rowA1 > n - 1) rowA1 = n - 1;
    const int rowP0 = pidx[rowA0];
    const int rowP1 = pidx[rowA1];
    const float* xA0 = x + (size_t)rowA0 * H;
    const float* xA1 = x + (size_t)rowA1 * H;
    const float* xP0 = x + (size_t)rowP0 * H;
    const float* xP1 = x + (size_t)rowP1 * H;

    v8f acc[MT][8] = {};
    for (int ks = 0; ks < 32; ++ks) {
        const int k0 = ks * 4 + half * 2;
        v2f aS0 = *(const v2f*)(xA0 + k0);
        v2f aS1 = *(const v2f*)(xA1 + k0);
        v2f aP0 = *(const v2f*)(xP0 + k0);
        v2f aP1 = *(const v2f*)(xP1 + k0);
#pragma unroll
        for (int nt = 0; nt < 8; ++nt) {
            const int col = nt * 16 + l;
            const float* wS = W_self + (size_t)k0 * H + col;
            const float* wP = W_par  + (size_t)k0 * H + col;
            v2f bS; bS.x = wS[0]; bS.y = wS[H];
            v2f bP; bP.x = wP[0]; bP.y = wP[H];
            acc[0][nt] = __builtin_amdgcn_wmma_f32_16x16x4_f32(
                false, aS0, false, bS, (short)0, acc[0][nt], false, false);
            acc[1][nt] = __builtin_amdgcn_wmma_f32_16x16x4_f32(
                false, aS1, false, bS, (short)0, acc[1][nt], false, false);
            acc[0][nt] = __builtin_amdgcn_wmma_f32_16x16x4_f32(
                false, aP0, false, bP, (short)0, acc[0][nt], false, false);
            acc[1][nt] = __builtin_amdgcn_wmma_f32_16x16x4_f32(
                false, aP1, false, bP, (short)0, acc[1][nt], false, false);
        }
    }

    // ---------------- Phase 2: acc += msg-sum (from LDS); write final agg back ----------------
#pragma unroll
    for (int t = 0; t < MT; ++t) {
#pragma unroll
        for (int nt = 0; nt < 8; ++nt) {
#pragma unroll
            for (int v = 0; v < 8; ++v) {
                const int rl = t * 16 + v + half * 8;
                acc[t][nt][v] += agg[rl * H + nt * 16 + l];
            }
        }
    }
#pragma unroll
    for (int t = 0; t < MT; ++t) {
#pragma unroll
        for (int nt = 0; nt < 8; ++nt) {
#pragma unroll
            for (int v = 0; v < 8; ++v) {
                const int rl = t * 16 + v + half * 8;
                agg[rl * H + nt * 16 + l] = acc[t][nt][v];
            }
        }
    }
    // DS ops from the same wave complete in order; the LDS tile is wave-private.

    // ---------------- Phase 3: out_tile = agg @ W_post + b_post ----------------
    v8f acc2[MT][8] = {};
    for (int ks = 0; ks < 32; ++ks) {
        const int k0 = ks * 4 + half * 2;
        v2f aA0 = *(const v2f*)(agg + l * H + k0);          // A-frag tile0 from LDS
        v2f aA1 = *(const v2f*)(agg + (16 + l) * H + k0);   // A-frag tile1 from LDS
#pragma unroll
        for (int nt = 0; nt < 8; ++nt) {
            const int col = nt * 16 + l;
            const float* wQ = W_post + (size_t)k0 * H + col;
            v2f bQ; bQ.x = wQ[0]; bQ.y = wQ[H];
            acc2[0][nt] = __builtin_amdgcn_wmma_f32_16x16x4_f32(
                false, aA0, false, bQ, (short)0, acc2[0][nt], false, false);
            acc2[1][nt] = __builtin_amdgcn_wmma_f32_16x16x4_f32(
                false, aA1, false, bQ, (short)0, acc2[1][nt], false, false);
        }
    }

    // ---------------- Phase 4: store ----------------
    float bp[8];
#pragma unroll
    for (int nt = 0; nt < 8; ++nt) bp[nt] = b_post[nt * 16 + l];

#pragma unroll
    for (int t = 0; t < MT; ++t) {
#pragma unroll
        for (int nt = 0; nt < 8; ++nt) {
#pragma unroll
            for (int v = 0; v < 8; ++v) {
                const int rl  = t * 16 + v + half * 8;
                const int row = r0 + rl;
                if (row < n) {
                    __builtin_nontemporal_store(acc2[t][nt][v] + bp[nt],
                                                out + (size_t)row * H + nt * 16 + l);
                }
            }
        }
    }
}

extern "C" void kernel_launch(void* const* d_in, const int* in_sizes, int n_in,
                              void* d_out, int out_size, void* d_ws, size_t ws_size,
                              hipStream_t stream) {
    const float*         x      = (const float*)d_in[0];
    const float*         msg    = (const float*)d_in[1];
    const unsigned char* mask   = (const unsigned char*)d_in[2];
    const int*           pidx   = (const int*)d_in[3];
    const float*         W_self = (const float*)d_in[4];
    const float*         b_self = (const float*)d_in[5];
    const float*         W_par  = (const float*)d_in[6];
    const float*         b_par  = (const float*)d_in[7];
    const float*         W_post = (const float*)d_in[8];
    const float*         b_post = (const float*)d_in[9];
    float* out = (float*)d_out;

    const int n     = in_sizes[0] / H;                  // 200000
    const int tiles = (n + ROWS_PER_WAVE - 1) / ROWS_PER_WAVE;
    const int grid  = (tiles + WAVES_PER_BLOCK - 1) / WAVES_PER_BLOCK;
    const size_t lds_bytes = (size_t)WAVES_PER_BLOCK * ROWS_PER_WAVE * H * sizeof(float); // 64 KB

    hipLaunchKernelGGL(chienn_fused_kernel, dim3(grid), dim3(WAVES_PER_BLOCK * 32),
                       lds_bytes, stream,
                       x, msg, mask, pidx, W_self, b_self, W_par, b_par,
                       W_post, b_post, out, n);
}